// IntentGraph_47957604827507
// MI455X (gfx1250) — compile-verified
//
#include <hip/hip_runtime.h>

#define H 128
#define NINT 512
#define LDP 132   // padded LDS row pitch (floats): 132 % 64 = 4 -> conflict-free row-strided reads
#define TOPK 3

typedef __attribute__((ext_vector_type(2))) float v2f;
typedef __attribute__((ext_vector_type(8))) float v8f;

__device__ __forceinline__ float warpSum(float v) {
#pragma unroll
  for (int off = 16; off > 0; off >>= 1) v += __shfl_xor(v, off, 32);
  return v;
}

// float atomic max via signed-int max / unsigned-int min punning
__device__ __forceinline__ void atomicMaxF(float* addr, float val) {
  if (val >= 0.0f) atomicMax((int*)addr, __float_as_int(val));
  else             atomicMin((unsigned int*)addr, __float_as_uint(val));
}

// ---------------------------------------------------------------------------
// init: m1 = -inf, d1 = 0, num = 0
__global__ void init_kernel(float* m1, float* d1, float* num) {
  int i = blockIdx.x * blockDim.x + threadIdx.x;
  if (i < NINT) { m1[i] = -3.0e38f; d1[i] = 0.0f; }
  if (i < NINT * H) num[i] = 0.0f;
}

// ---------------------------------------------------------------------------
// K' = intent_emb @ Wk^T   [512 x 128] (tiny)
__global__ __launch_bounds__(H) void kp_kernel(const float* __restrict__ intent,
                                               const float* __restrict__ Wk,
                                               float* __restrict__ Kp) {
  __shared__ float row[H];
  const int j = blockIdx.x, n = threadIdx.x;
  row[n] = intent[(size_t)j * H + n];
  __syncthreads();
  float acc = 0.0f;
#pragma unroll 8
  for (int h = 0; h < H; ++h) acc += row[h] * Wk[(size_t)n * H + h];
  Kp[(size_t)j * H + n] = acc;
}

// ---------------------------------------------------------------------------
// Fused: Q = item@Wq^T ; sim = Q@K'^T ; per-row top-3 (softmax is monotone -> skip)
// block = 128 threads (4 waves), 32 items per wave -> 128 items/block.
// fp32 WMMA 16x16x4, A fragment held in registers, dual accumulators share B.
__global__ __launch_bounds__(128) void sim_topk_kernel(
    const float* __restrict__ item, const float* __restrict__ Wq,
    const float* __restrict__ Kp, int* __restrict__ topi, int nItems) {
  extern __shared__ float lds[];
  float* ldsItem = lds;               // 128 x LDP  (reused for per-wave sim tiles in phase 2)
  float* ldsWq   = lds + H * LDP;     // 128 x LDP  (reused for K' chunk, 64 x LDP, in phase 2)
  float* ldsQ    = lds + 2 * H * LDP; // 128 x LDP
  float* ldsKp   = ldsWq;
  float* simBase = ldsItem;           // 4 waves * 2048 floats

  const int tid  = threadIdx.x;
  const int wave = tid >> 5;          // 0..3
  const int lane = tid & 31;
  const int ln15 = lane & 15;
  const int koff = (lane >> 4) * 2;   // A/B operand K offset for upper half-wave
  const int hi8  = (lane >> 4) * 8;   // C/D row offset for upper half-wave
  const int itemBase = blockIdx.x * 128;

  // Stage Wq and the 128-item tile (coalesced b128 loads, padded rows).
  for (int idx = tid; idx < H * (H / 4); idx += 128) {
    const int row = idx >> 5, cv = idx & 31;
    ((float4*)(ldsWq + row * LDP))[cv] = ((const float4*)(Wq + (size_t)row * H))[cv];
    const int g = itemBase + row;
    float4 v;
    if (g < nItems) v = ((const float4*)(item + (size_t)g * H))[cv];
    else { v.x = v.y = v.z = v.w = 0.0f; }
    ((float4*)(ldsItem + row * LDP))[cv] = v;
  }
  __syncthreads();

  // A fragments for this wave's two 16-row groups, all 32 K-steps, in registers.
  const int rowA = wave * 32 + ln15;   // rows rowA and rowA+16
  v2f aq0[32], aq1[32];
#pragma unroll
  for (int k = 0; k < 32; ++k) {
    aq0[k].x = ldsItem[rowA * LDP + k * 4 + koff];
    aq0[k].y = ldsItem[rowA * LDP + k * 4 + koff + 1];
    aq1[k].x = ldsItem[(rowA + 16) * LDP + k * 4 + koff];
    aq1[k].y = ldsItem[(rowA + 16) * LDP + k * 4 + koff + 1];
  }

  // Phase 1: Q = item @ Wq^T (two 16x16 tiles per n-tile, shared B fragment)
  for (int nt = 0; nt < 8; ++nt) {
    v8f c0 = {}, c1 = {};
    const int col = nt * 16 + ln15;
#pragma unroll
    for (int k = 0; k < 32; ++k) {
      v2f b;
      b.x = ldsWq[col * LDP + k * 4 + koff];   // B[k][n] = Wq[n][k]
      b.y = ldsWq[col * LDP + k * 4 + koff + 1];
      c0 = __builtin_amdgcn_wmma_f32_16x16x4_f32(false, aq0[k], false, b, (short)0, c0, false, false);
      c1 = __builtin_amdgcn_wmma_f32_16x16x4_f32(false, aq1[k], false, b, (short)0, c1, false, false);
    }
#pragma unroll
    for (int r = 0; r < 8; ++r) {
      ldsQ[(wave * 32 + hi8 + r) * LDP + nt * 16 + ln15]      = c0[r];
      ldsQ[(wave * 32 + 16 + hi8 + r) * LDP + nt * 16 + ln15] = c1[r];
    }
  }

  // Reload A fragments as Q (same wave wrote its own rows; DS is in-order per wave).
#pragma unroll
  for (int k = 0; k < 32; ++k) {
    aq0[k].x = ldsQ[rowA * LDP + k * 4 + koff];
    aq0[k].y = ldsQ[rowA * LDP + k * 4 + koff + 1];
    aq1[k].x = ldsQ[(rowA + 16) * LDP + k * 4 + koff];
    aq1[k].y = ldsQ[(rowA + 16) * LDP + k * 4 + koff + 1];
  }

  // Phase 2: sim strip (32 x 512) in 64-intent chunks; streaming top-3.
  float t0 = -3.0e38f, t1 = -3.0e38f, t2 = -3.0e38f;
  int   i0 = 0, i1 = 1, i2 = 2;
  float* simBuf = simBase + wave * 2048;          // 4 tiles x (2 x 256 floats)
  const int scanSel = (lane >> 4) * 256;          // which 16-row half this lane scans
  for (int chunk = 0; chunk < NINT / 64; ++chunk) {
    __syncthreads();
    for (int idx = tid; idx < 64 * (H / 4); idx += 128) {
      const int row = idx >> 5, cv = idx & 31;
      ((float4*)(ldsKp + row * LDP))[cv] =
          ((const float4*)(Kp + (size_t)(chunk * 64 + row) * H))[cv];
    }
    if (chunk + 1 < NINT / 64 && tid < 64)  // global_prefetch_b8 next chunk
      __builtin_prefetch(Kp + (size_t)((chunk + 1) * 64 + tid) * H, 0, 0);
    __syncthreads();
    for (int jt = 0; jt < 4; ++jt) {
      v8f c0 = {}, c1 = {};
      const int jl = jt * 16 + ln15;
#pragma unroll
      for (int k = 0; k < 32; ++k) {
        v2f b;
        b.x = ldsKp[jl * LDP + k * 4 + koff];    // B[k][j] = Kp[j][k]
        b.y = ldsKp[jl * LDP + k * 4 + koff + 1];
        c0 = __builtin_amdgcn_wmma_f32_16x16x4_f32(false, aq0[k], false, b, (short)0, c0, false, false);
        c1 = __builtin_amdgcn_wmma_f32_16x16x4_f32(false, aq1[k], false, b, (short)0, c1, false, false);
      }
#pragma unroll
      for (int r = 0; r < 8; ++r) {
        const int idx = (hi8 + r) * 16 + ln15;
        simBuf[jt * 512 + idx]       = c0[r];
        simBuf[jt * 512 + 256 + idx] = c1[r];
      }
    }
    __syncthreads();   // make this wave's tile writes visible across its lanes
    {
#pragma unroll
      for (int jt = 0; jt < 4; ++jt) {
        const int base = jt * 512 + scanSel + ln15 * 16;
        const int jbase = chunk * 64 + jt * 16;
#pragma unroll
        for (int n = 0; n < 16; ++n) {
          const float v = simBuf[base + n];
          const int j = jbase + n;
          if (v > t0)      { t2 = t1; i2 = i1; t1 = t0; i1 = i0; t0 = v; i0 = j; }
          else if (v > t1) { t2 = t1; i2 = i1; t1 = v;  i1 = j; }
          else if (v > t2) { t2 = v;  i2 = j; }
        }
      }
    }
  }
  const int g = itemBase + wave * 32 + lane;   // every lane owns one row
  if (g < nItems) { topi[g * 3 + 0] = i0; topi[g * 3 + 1] = i1; topi[g * 3 + 2] = i2; }
}

// ---------------------------------------------------------------------------
// GNN pass A: s1 per edge + segment max (wave per edge)
__global__ __launch_bounds__(256) void edge_s1_kernel(
    const float* __restrict__ item, const float* __restrict__ intent,
    const float* __restrict__ wa, const int* __restrict__ topi,
    float* __restrict__ s1, float* __restrict__ m1, int nEdges) {
  const int e = blockIdx.x * 8 + (threadIdx.x >> 5);
  const int lane = threadIdx.x & 31;
  if (e >= nEdges) return;
  const int it = e / 3;
  const int ei = topi[e];
  const float* ir = item + (size_t)it * H;
  const float* nr = intent + (size_t)ei * H;
  float acc = 0.0f;
#pragma unroll
  for (int h = lane; h < H; h += 32) acc += ir[h] * nr[h] * wa[h];
  acc = warpSum(acc);
  const float s = acc > 0.0f ? acc : 0.2f * acc;   // LeakyReLU(0.2)
  if (lane == 0) { s1[e] = s; atomicMaxF(&m1[ei], s); }
}

// GNN pass B: e1 = exp(s1-m1), accumulate denominator and numerator
__global__ __launch_bounds__(256) void edge_acc_kernel(
    const float* __restrict__ item, const float* __restrict__ s1,
    const float* __restrict__ m1, const int* __restrict__ topi,
    float* __restrict__ d1, float* __restrict__ num, int nEdges) {
  const int e = blockIdx.x * 8 + (threadIdx.x >> 5);
  const int lane = threadIdx.x & 31;
  if (e >= nEdges) return;
  const int it = e / 3;
  const int ei = topi[e];
  const float w = __expf(s1[e] - m1[ei]);
  if (lane == 0) atomicAdd(&d1[ei], w);
  const float* ir = item + (size_t)it * H;
  float* nr = num + (size_t)ei * H;
#pragma unroll
  for (int h = lane; h < H; h += 32) atomicAdd(&nr[h], w * ir[h]);
}

// intent_new = num / d1
__global__ void inew_kernel(const float* __restrict__ num,
                            const float* __restrict__ d1,
                            float* __restrict__ inew) {
  const int i = blockIdx.x * blockDim.x + threadIdx.x;
  if (i < NINT * H) inew[i] = num[i] / d1[i >> 7];
}

// att2 + blend: wave per item, softmax over its 3 edges
__global__ __launch_bounds__(256) void final_kernel(
    const float* __restrict__ item, const float* __restrict__ inew,
    const float* __restrict__ wb, const int* __restrict__ topi,
    float* __restrict__ out, int nItems) {
  const int i = blockIdx.x * 8 + (threadIdx.x >> 5);
  const int lane = threadIdx.x & 31;
  if (i >= nItems) return;
  const int j0 = topi[i * 3 + 0], j1 = topi[i * 3 + 1], j2 = topi[i * 3 + 2];
  const float* ir = item + (size_t)i * H;
  const float* r0 = inew + (size_t)j0 * H;
  const float* r1 = inew + (size_t)j1 * H;
  const float* r2 = inew + (size_t)j2 * H;
  float a0 = 0.0f, a1 = 0.0f, a2 = 0.0f;
#pragma unroll
  for (int h = lane; h < H; h += 32) {
    const float x = ir[h] * wb[h];
    a0 += x * r0[h]; a1 += x * r1[h]; a2 += x * r2[h];
  }
  a0 = warpSum(a0); a1 = warpSum(a1); a2 = warpSum(a2);
  a0 = a0 > 0.0f ? a0 : 0.2f * a0;
  a1 = a1 > 0.0f ? a1 : 0.2f * a1;
  a2 = a2 > 0.0f ? a2 : 0.2f * a2;
  const float m = fmaxf(a0, fmaxf(a1, a2));
  float w0 = __expf(a0 - m), w1 = __expf(a1 - m), w2 = __expf(a2 - m);
  const float inv = 1.0f / (w0 + w1 + w2);
  w0 *= inv; w1 *= inv; w2 *= inv;
#pragma unroll
  for (int h = lane; h < H; h += 32)
    out[(size_t)i * H + h] = 0.5f * ir[h] + 0.5f * (w0 * r0[h] + w1 * r1[h] + w2 * r2[h]);
}

// ---------------------------------------------------------------------------
extern "C" void kernel_launch(void* const* d_in, const int* in_sizes, int n_in,
                              void* d_out, int out_size, void* d_ws, size_t ws_size,
                              hipStream_t stream) {
  const float* item   = (const float*)d_in[0];
  const float* intent = (const float*)d_in[1];
  const float* Wq     = (const float*)d_in[2];
  const float* Wk     = (const float*)d_in[3];
  const float* wa     = (const float*)d_in[4];
  const float* wb     = (const float*)d_in[5];
  const int nItems = in_sizes[0] / H;
  const int nEdges = nItems * TOPK;
  float* out = (float*)d_out;

  // workspace layout
  float* ws   = (float*)d_ws;
  float* Kp   = ws;                               // 512*128
  float* s1   = Kp + NINT * H;                    // nEdges (padded)
  float* m1   = s1 + ((nEdges + 31) & ~31);       // 512
  float* d1v  = m1 + NINT;                        // 512
  float* num  = d1v + NINT;                       // 512*128
  float* inw  = num + NINT * H;                   // 512*128
  int*   topi = (int*)(inw + NINT * H);           // nEdges ints

  init_kernel<<<(NINT * H + 255) / 256, 256, 0, stream>>>(m1, d1v, num);
  kp_kernel<<<NINT, H, 0, stream>>>(intent, Wk, Kp);

  const size_t ldsBytes = (size_t)3 * H * LDP * sizeof(float);  // ~198KB dynamic LDS (WGP has 320KB)
  sim_topk_kernel<<<(nItems + 127) / 128, 128, ldsBytes, stream>>>(item, Wq, Kp, topi, nItems);

  edge_s1_kernel<<<(nEdges + 7) / 8, 256, 0, stream>>>(item, intent, wa, topi, s1, m1, nEdges);
  edge_acc_kernel<<<(nEdges + 7) / 8, 256, 0, stream>>>(item, s1, m1, topi, d1v, num, nEdges);
  inew_kernel<<<(NINT * H + 255) / 256, 256, 0, stream>>>(num, d1v, inw);
  final_kernel<<<(nItems + 7) / 8, 256, 0, stream>>>(item, inw, wb, topi, out, nItems);
}